// QuantumReproductiveTensor_25967372271627
// MI455X (gfx1250) — compile-verified
//
#include <hip/hip_runtime.h>

// ---------------------------------------------------------------------------
// QuantumReproductiveTensor on MI455X (gfx1250)
//
// out[k] = (G @ (0.6*ps_n + 0.4*(L @ pb_n)))[k] + 0.1 * Re(einsum kij,i,j->k)
// Dominant cost: streaming the 512^3 complex64 tensor T (1.07 GB) once.
// Memory-bound: ~46 us floor at 23.3 TB/s. fp32 WMMA (16x16x4) drives the
// big reduction via a diagonal-dot formulation with fully coalesced loads.
// ---------------------------------------------------------------------------

#define D 512
#define KFLOATS (2 * D * D)   // 524288 floats per output row of T

typedef float v2f __attribute__((ext_vector_type(2)));
typedef float v8f __attribute__((ext_vector_type(8)));

static __device__ __forceinline__ v2f ld_nt_v2(const float* p) {
    return __builtin_nontemporal_load((const v2f*)p);
}
static __device__ __forceinline__ v2f ld_v2(const float* p) {
    return *(const v2f*)p;
}
static __device__ __forceinline__ v8f wmma_f32(v2f a, v2f b, v8f c) {
    // D = A(16x4) x B(4x16) + C ; 8-arg f32 pattern:
    // (neg_a, A, neg_b, B, c_mod, C, reuse_a, reuse_b)
    return __builtin_amdgcn_wmma_f32_16x16x4_f32(false, a, false, b,
                                                 (short)0, c, false, false);
}

// ---- Kernel A: norms + normalized vectors (complex treated as 1024 floats)
__global__ __launch_bounds__(256) void k_norms(const float* __restrict__ ps,
                                               const float* __restrict__ pb,
                                               float* __restrict__ ps_n,
                                               float* __restrict__ pb_n) {
    __shared__ float s1[256], s2[256];
    int t = threadIdx.x;
    float a = 0.f, b = 0.f;
    for (int i = t; i < 2 * D; i += 256) {
        float x = ps[i]; a += x * x;
        float y = pb[i]; b += y * y;
    }
    s1[t] = a; s2[t] = b;
    __syncthreads();
    for (int o = 128; o; o >>= 1) {
        if (t < o) { s1[t] += s1[t + o]; s2[t] += s2[t + o]; }
        __syncthreads();
    }
    float ia = 1.0f / sqrtf(s1[0]);
    float ib = 1.0f / sqrtf(s2[0]);
    for (int i = t; i < 2 * D; i += 256) {
        ps_n[i] = ps[i] * ia;
        pb_n[i] = pb[i] * ib;
    }
}

// ---- Kernel B: W'[2m] = Re(ps_i * pb_j), W'[2m+1] = -Im(ps_i * pb_j)
__global__ __launch_bounds__(256) void k_outer(const float* __restrict__ ps_n,
                                               const float* __restrict__ pb_n,
                                               float* __restrict__ wp) {
    int i = blockIdx.x;
    float pr = ps_n[2 * i], pi = ps_n[2 * i + 1];
    float2* w2 = (float2*)wp;
    for (int j = threadIdx.x; j < D; j += 256) {
        float br = pb_n[2 * j], bi = pb_n[2 * j + 1];
        float2 v;
        v.x =  pr * br - pi * bi;   //  Re(ps*pb)
        v.y = -(pr * bi + pi * br); // -Im(ps*pb)
        w2[i * D + j] = v;
    }
}

// ---- Kernel C: emergence_real[k] = dot(T_row_k, W')  (K = 524288 floats)
// One block per k; 16 waves; each wave owns 64-float chunks strided by 1024.
// WMMA diagonal-dot: A[m,t]=Tc[4m+t], B[t,j]=Wc[4j+t]  =>  sum(diag D) = dot.
__global__ __launch_bounds__(512) void k_emerge(const float* __restrict__ T,
                                                const float* __restrict__ wp,
                                                float* __restrict__ em) {
    const int k    = blockIdx.x;
    const int tid  = threadIdx.x;
    const int lane = tid & 31;
    const int wave = tid >> 5;                       // 0..15
    // Per-lane float offset shared by A and B operand layouts:
    // lane l<16 -> 4*l + {0,1} ; lane l>=16 -> 4*(l-16) + {2,3}
    const int laneoff = ((lane & 15) << 2) | ((lane >> 4) << 1);

    const float* tp = T + (size_t)k * KFLOATS + wave * 64 + laneoff;
    const float* qp = wp + wave * 64 + laneoff;

    v8f acc0 = {};
    v8f acc1 = {};
    // 512 chunks per wave, unrolled x4 (immediate offsets 0/4KB/8KB/12KB)
    for (int t = 0; t < 512; t += 4) {
        v2f a0 = ld_nt_v2(tp);
        v2f a1 = ld_nt_v2(tp + 1024);
        v2f a2 = ld_nt_v2(tp + 2048);
        v2f a3 = ld_nt_v2(tp + 3072);
        v2f b0 = ld_v2(qp);
        v2f b1 = ld_v2(qp + 1024);
        v2f b2 = ld_v2(qp + 2048);
        v2f b3 = ld_v2(qp + 3072);
        acc0 = wmma_f32(a0, b0, acc0);
        acc1 = wmma_f32(a1, b1, acc1);
        acc0 = wmma_f32(a2, b2, acc0);
        acc1 = wmma_f32(a3, b3, acc1);
        tp += 4096;
        qp += 4096;
    }
    v8f acc = acc0 + acc1;

    // Extract diagonal of the 16x16 accumulator:
    // D[r,r]   lives in VGPR r at lane r      (r = 0..7)
    // D[r+8,r+8] lives in VGPR r at lane r+24 (r = 0..7)
    float part = 0.f;
#pragma unroll
    for (int r = 0; r < 8; ++r) {
        if (lane == r)      part = acc[r];
        if (lane == r + 24) part = acc[r];
    }
    // wave32 reduce
    for (int off = 16; off; off >>= 1)
        part += __shfl_down(part, off, 32);

    __shared__ float sred[16];
    if (lane == 0) sred[wave] = part;
    __syncthreads();
    if (tid == 0) {
        float v = 0.f;
#pragma unroll
        for (int i = 0; i < 16; ++i) v += sred[i];
        em[k] = v;
    }
}

// ---- Kernel D1: child_base = 0.6*ps_n + 0.4*(L @ pb_n)   (complex matvec)
__global__ __launch_bounds__(256) void k_love(const float* __restrict__ L,
                                              const float* __restrict__ ps_n,
                                              const float* __restrict__ pb_n,
                                              float* __restrict__ cb) {
    int row  = blockIdx.x * 8 + (threadIdx.x >> 5);  // 64 blocks * 8 waves
    int lane = threadIdx.x & 31;
    const float* lr = L + (size_t)row * (2 * D);
    float cr = 0.f, ci = 0.f;
    for (int j = lane; j < D; j += 32) {
        float ar = lr[2 * j], ai = lr[2 * j + 1];
        float br = pb_n[2 * j], bi = pb_n[2 * j + 1];
        cr += ar * br - ai * bi;
        ci += ar * bi + ai * br;
    }
    for (int off = 16; off; off >>= 1) {
        cr += __shfl_down(cr, off, 32);
        ci += __shfl_down(ci, off, 32);
    }
    if (lane == 0) {
        cb[2 * row]     = 0.6f * ps_n[2 * row]     + 0.4f * cr;
        cb[2 * row + 1] = 0.6f * ps_n[2 * row + 1] + 0.4f * ci;
    }
}

// ---- Kernel D2: out = G @ child_base + 0.1*emergence_real (into real part)
__global__ __launch_bounds__(256) void k_child(const float* __restrict__ G,
                                               const float* __restrict__ cb,
                                               const float* __restrict__ em,
                                               float* __restrict__ out) {
    int row  = blockIdx.x * 8 + (threadIdx.x >> 5);
    int lane = threadIdx.x & 31;
    const float* gr = G + (size_t)row * (2 * D);
    float cr = 0.f, ci = 0.f;
    for (int j = lane; j < D; j += 32) {
        float ar = gr[2 * j], ai = gr[2 * j + 1];
        float br = cb[2 * j], bi = cb[2 * j + 1];
        cr += ar * br - ai * bi;
        ci += ar * bi + ai * br;
    }
    for (int off = 16; off; off >>= 1) {
        cr += __shfl_down(cr, off, 32);
        ci += __shfl_down(ci, off, 32);
    }
    if (lane == 0) {
        out[2 * row]     = cr + 0.1f * em[row];
        out[2 * row + 1] = ci;
    }
}

extern "C" void kernel_launch(void* const* d_in, const int* in_sizes, int n_in,
                              void* d_out, int out_size, void* d_ws, size_t ws_size,
                              hipStream_t stream) {
    const float* ps = (const float*)d_in[0];  // parent_self      complex64[512]
    const float* pb = (const float*)d_in[1];  // parent_beloved   complex64[512]
    const float* L  = (const float*)d_in[2];  // love_coupling    complex64[512,512]
    const float* G  = (const float*)d_in[3];  // genetic_operator complex64[512,512]
    const float* T  = (const float*)d_in[4];  // creative_emergence complex64[512^3]
    float* out = (float*)d_out;               // complex64[512] -> 1024 floats

    float* ws   = (float*)d_ws;
    float* ps_n = ws;                         // 1024 floats
    float* pb_n = ws + 1024;                  // 1024 floats
    float* wp   = ws + 2048;                  // 524288 floats (W')
    float* em   = wp + KFLOATS;               // 512 floats
    float* cb   = em + D;                     // 1024 floats (child_base)

    k_norms <<<1,   256, 0, stream>>>(ps, pb, ps_n, pb_n);
    k_outer <<<D,   256, 0, stream>>>(ps_n, pb_n, wp);
    k_love  <<<64,  256, 0, stream>>>(L, ps_n, pb_n, cb);
    k_emerge<<<D,   512, 0, stream>>>(T, wp, em);
    k_child <<<64,  256, 0, stream>>>(G, cb, em, out);
}